// RegionProposalNetwork_69758858822043
// MI455X (gfx1250) — compile-verified
//
#include <hip/hip_runtime.h>
#include <math.h>

typedef float v2f __attribute__((ext_vector_type(2)));
typedef float v8f __attribute__((ext_vector_type(8)));

#define C_DIM   512
#define HW      6144            // 64*96 pixels per image
#define NIMG    16
#define O_LOC   36
#define O_SC    18
#define O_TOT   54
#define O_PAD   64              // 4 N-tiles of 16
#define KCHUNK  128             // K staged into LDS per stage (32 KB)
#define PAIR_STRIDE (O_PAD * 2) // floats per K-pair row in LDS

#define M_TOTAL   (NIMG * HW)          // 98304 pixels
#define TILES     (M_TOTAL / 16)       // 6144 M-tiles of 16 pixels
#define TILES_PER_IMG (HW / 16)        // 384 (tiles never cross images)

#define LOC_ELEMS ((size_t)NIMG * HW * O_LOC)   // 3,538,944
#define SC_ELEMS  ((size_t)NIMG * HW * O_SC)    // 1,769,472
#define FG_ELEMS  ((size_t)NIMG * HW * 9)       // 884,736

// ---------------------------------------------------------------------------
// Main RPN head: fused loc+score GEMM (FP32 WMMA 16x16x4) + bias + softmax-fg
// Block = 256 threads = 8 waves. Each wave: 2 M-tiles x 4 N-tiles, K=512.
// LDS weight layout is K-pair interleaved: (k,o) -> sW[(k>>1)*128 + o*2 + (k&1)]
// so every B fragment is a single aligned ds_load_b64 into a WMMA reg pair.
// ---------------------------------------------------------------------------
__global__ __launch_bounds__(256)
void rpn_head_kernel(const float* __restrict__ x,
                     const float* __restrict__ loc_w,
                     const float* __restrict__ loc_b,
                     const float* __restrict__ score_w,
                     const float* __restrict__ score_b,
                     float* __restrict__ d_out)
{
    __shared__ float sW[(KCHUNK / 2) * PAIR_STRIDE];  // 32 KB
    __shared__ float sBias[O_PAD];

    const int tid  = threadIdx.x;
    const int wave = tid >> 5;
    const int lane = tid & 31;
    const int col  = lane & 15;            // N / M-row selector within half-wave
    const int khalf = lane >> 4;           // 0: K={0,1}, 1: K={2,3} of each step

    // stage biases once (padded with zeros)
    if (tid < O_PAD) {
        float b = 0.0f;
        if (tid < O_LOC)       b = loc_b[tid];
        else if (tid < O_TOT)  b = score_b[tid - O_LOC];
        sBias[tid] = b;
    }

    // two M-tiles per wave
    const int tile0 = blockIdx.x * 16 + wave * 2;
    const int tile1 = tile0 + 1;

    const int n0 = tile0 / TILES_PER_IMG;
    const int n1 = tile1 / TILES_PER_IMG;
    const size_t base0 = (size_t)n0 * C_DIM * HW + (size_t)(tile0 % TILES_PER_IMG) * 16 + col;
    const size_t base1 = (size_t)n1 * C_DIM * HW + (size_t)(tile1 % TILES_PER_IMG) * 16 + col;

    v8f acc[2][4];
    const v8f vzero = {0.f,0.f,0.f,0.f,0.f,0.f,0.f,0.f};
    #pragma unroll
    for (int t = 0; t < 2; ++t)
        #pragma unroll
        for (int j = 0; j < 4; ++j)
            acc[t][j] = vzero;

    const int kq = khalf << 1;             // K sub-offset for this half-wave

    for (int kc = 0; kc < C_DIM / KCHUNK; ++kc) {
        const int k0 = kc * KCHUNK;

        __syncthreads();   // previous chunk fully consumed
        // stage W^T chunk into LDS, K-pair interleaved
        for (int idx = tid; idx < O_TOT * KCHUNK; idx += 256) {
            const int o  = idx >> 7;        // / KCHUNK
            const int kk = idx & (KCHUNK - 1);
            const float wv = (o < O_LOC)
                ? loc_w[o * C_DIM + k0 + kk]
                : score_w[(o - O_LOC) * C_DIM + k0 + kk];
            sW[(kk >> 1) * PAIR_STRIDE + o * 2 + (kk & 1)] = wv;
        }
        __syncthreads();

        for (int ks = 0; ks < KCHUNK; ks += 4) {
            const int c0 = k0 + ks + kq;

            // A fragments: 16x4 f32, per-lane {K=kq, K=kq+1} for row M=col
            v2f a0, a1;
            a0.x = x[base0 + (size_t)(c0    ) * HW];
            a0.y = x[base0 + (size_t)(c0 + 1) * HW];
            a1.x = x[base1 + (size_t)(c0    ) * HW];
            a1.y = x[base1 + (size_t)(c0 + 1) * HW];

            // B fragments: one aligned b64 LDS load each, no marshalling
            const int pairRow = (ks >> 1) + khalf;
            const float* pb = &sW[pairRow * PAIR_STRIDE + col * 2];
            v2f b[4];
            #pragma unroll
            for (int j = 0; j < 4; ++j)
                b[j] = *(const v2f*)(pb + j * 32);

            #pragma unroll
            for (int j = 0; j < 4; ++j) {
                acc[0][j] = __builtin_amdgcn_wmma_f32_16x16x4_f32(
                    false, a0, false, b[j], (short)0, acc[0][j], false, false);
                acc[1][j] = __builtin_amdgcn_wmma_f32_16x16x4_f32(
                    false, a1, false, b[j], (short)0, acc[1][j], false, false);
            }

            // prefetch next K-slab of x (emits global_prefetch_b8)
            if (ks + 4 < KCHUNK) {
                __builtin_prefetch((const void*)&x[base0 + (size_t)(c0 + 4) * HW], 0, 1);
                __builtin_prefetch((const void*)&x[base1 + (size_t)(c0 + 4) * HW], 0, 1);
            }
        }
    }

    // ---------------- epilogue: bias, scatter, fg = sigmoid(s1 - s0) --------
    float* out_loc   = d_out;
    float* out_score = d_out + LOC_ELEMS;
    float* out_fg    = d_out + LOC_ELEMS + SC_ELEMS;

    const int mhi = khalf * 8;             // C/D layout: lanes 16-31 hold M+8

    #pragma unroll
    for (int t = 0; t < 2; ++t) {
        const int tile = (t == 0) ? tile0 : tile1;
        #pragma unroll
        for (int j = 0; j < 4; ++j) {
            const int o = j * 16 + col;
            const float bias = sBias[o];
            #pragma unroll
            for (int v = 0; v < 8; ++v) {
                const float val = acc[t][j][v] + bias;
                // partner lane holds the other logit of the softmax pair
                const float partner = __shfl_xor(val, 1, 32);
                const size_t p = (size_t)tile * 16 + (size_t)(v + mhi);
                if (o < O_LOC) {
                    out_loc[p * O_LOC + o] = val;
                } else if (o < O_TOT) {
                    const int s = o - O_LOC;
                    out_score[p * O_SC + s] = val;
                    if ((s & 1) == 0) {
                        // fg = exp(s1)/(exp(s0)+exp(s1)) = sigmoid(s1 - s0)
                        const float fg = 1.0f / (1.0f + __expf(val - partner));
                        out_fg[p * 9 + (s >> 1)] = fg;
                    }
                }
            }
        }
    }
}

// ---------------------------------------------------------------------------
// Anchor table: (hh*ww*9, 4) in (y1,x1,y2,x2), n-independent
// ---------------------------------------------------------------------------
__global__ __launch_bounds__(256)
void rpn_anchor_kernel(float* __restrict__ out_anchor)
{
    const int i = blockIdx.x * 256 + threadIdx.x;   // grid sized exactly
    const int hw = i / 9;
    const int a  = i - hw * 9;
    const int h  = hw / 96;
    const int w  = hw - h * 96;
    const int ri = a / 3;
    const int si = a - ri * 3;

    const float ratios[3] = {0.5f, 1.0f, 2.0f};
    const float scales[3] = {8.0f, 16.0f, 32.0f};
    const float r = ratios[ri];
    const float s = scales[si];

    const float hbox = 16.0f * s * __fsqrt_rn(r);
    const float wbox = 16.0f * s * __fsqrt_rn(1.0f / r);
    const float py = (float)h * 16.0f + 8.0f;
    const float px = (float)w * 16.0f + 8.0f;

    float4 v;
    v.x = py - hbox * 0.5f;
    v.y = px - wbox * 0.5f;
    v.z = py + hbox * 0.5f;
    v.w = px + wbox * 0.5f;
    ((float4*)out_anchor)[i] = v;
}

// ---------------------------------------------------------------------------
extern "C" void kernel_launch(void* const* d_in, const int* in_sizes, int n_in,
                              void* d_out, int out_size, void* d_ws, size_t ws_size,
                              hipStream_t stream)
{
    const float* x       = (const float*)d_in[0];
    const float* loc_w   = (const float*)d_in[1];
    const float* loc_b   = (const float*)d_in[2];
    const float* score_w = (const float*)d_in[3];
    const float* score_b = (const float*)d_in[4];
    float* out = (float*)d_out;

    // GEMM head: 6144 M-tiles / (8 waves * 2 tiles) = 384 blocks (exact)
    rpn_head_kernel<<<TILES / 16, 256, 0, stream>>>(x, loc_w, loc_b, score_w, score_b, out);

    // anchors: 55296 anchors / 256 = 216 blocks (exact)
    float* out_anchor = out + LOC_ELEMS + SC_ELEMS + FG_ELEMS;
    rpn_anchor_kernel<<<(HW * 9) / 256, 256, 0, stream>>>(out_anchor);
}